// BasicSoftmaxRouter_8083128451222
// MI455X (gfx1250) — compile-verified
//
#include <hip/hip_runtime.h>
#include <hip/hip_bf16.h>
#include <math.h>

// ---------------------------------------------------------------------------
// Noisy top-k router for MI455X (gfx1250, wave32).
//   logits[t][e]   = sum_d x[t][d] * w_g[e][d]
//   noisepre[t][e] = sum_d x[t][d] * w_noise[e][d]
//   final[t][e]    = logits + softplus(noisepre) * eps[t][e]
//   out = top2 values (f32) ++ top2 indices (i32)
//
// Fused dual-GEMV via V_WMMA_F32_16X16X4_F32 (B cols 0..7 = w_g, 8..15 =
// w_noise). x is streamed through LDS with double-buffered
// GLOBAL_LOAD_ASYNC_TO_LDS_B128 copies (ASYNCcnt pipeline, 16 KB in flight
// per wave) so HBM (23.3 TB/s) can be saturated even at 1 wave/SIMD.
// ---------------------------------------------------------------------------

typedef __attribute__((ext_vector_type(2))) float v2f;
typedef __attribute__((ext_vector_type(8))) float v8f;

#define RD 2048            // model dim D
#define RE 8               // experts
#define RTOK 16384         // B*S tokens
#define KB 128             // K-block floats per stage (512 B per row)
#define NSTAGES (RD / KB)  // 16
#define CPS (KB / 4)       // 32 wmma chunks per stage
#define WAVES_PER_BLOCK 4

__device__ __forceinline__ unsigned lds_off(const void* p) {
    // low 32 bits of a generic pointer to LDS == wave-relative LDS byte addr
    return (unsigned)(uintptr_t)p;
}

__device__ __forceinline__ void async_stage(uint64_t xbase, unsigned gvoff,
                                            unsigned ldsbase, int lane) {
    // Stage 16 rows x 512 B: one fully-coalesced b128 async copy per row.
    #pragma unroll
    for (int r = 0; r < 16; ++r) {
        unsigned l = ldsbase + (unsigned)(r * 512) + (unsigned)(lane * 16);
        unsigned g = gvoff + (unsigned)(r * 8192) + (unsigned)(lane * 16);
        asm volatile("global_load_async_to_lds_b128 %0, %1, %2"
                     :: "v"(l), "v"(g), "s"(xbase)
                     : "memory");
    }
}

__device__ __forceinline__ void wait_async_le16() {
    asm volatile("s_wait_asynccnt 0x10" ::: "memory");
}
__device__ __forceinline__ void wait_async_0() {
    asm volatile("s_wait_asynccnt 0x0" ::: "memory");
}
__device__ __forceinline__ void wait_ds_0() {
    asm volatile("s_wait_dscnt 0x0" ::: "memory");
}

__global__ __launch_bounds__(128) void noisy_topk_router_wmma(
    const float* __restrict__ x,        // [RTOK][RD]
    const float* __restrict__ w_g,      // [RE][RD]
    const float* __restrict__ w_noise,  // [RE][RD]
    const float* __restrict__ eps,      // [RTOK][RE]
    float* __restrict__ out)            // [RTOK*2] f32 ++ [RTOK*2] i32
{
    __shared__ __align__(16) float xstage[WAVES_PER_BLOCK][2][16][KB]; // 64 KB
    __shared__ __align__(64) float cscr[WAVES_PER_BLOCK][16][16];      //  4 KB

    const int tid  = threadIdx.x;
    const int wave = tid >> 5;
    const int lane = tid & 31;
    const int m    = lane & 15;   // A row / B column index
    const int hi   = lane >> 4;   // 0: K={0,1}, 1: K={2,3} within a chunk

    const int tile = blockIdx.x * WAVES_PER_BLOCK + wave;   // 16-token tile

    // --- async x staging setup -------------------------------------------
    const uint64_t xbase   = (uint64_t)x;
    const unsigned tileoff = (unsigned)tile * (unsigned)(16 * RD * 4); // bytes
    const unsigned ldsb0   = lds_off(&xstage[wave][0][0][0]);
    const unsigned ldsb1   = lds_off(&xstage[wave][1][0][0]);

    // --- B fragment source: column n=m -> w_g row m / w_noise row m-8 ----
    const float* wrow = (m < 8) ? (w_g + (size_t)m * RD)
                                : (w_noise + (size_t)(m - 8) * RD);
    const v2f* bp = (const v2f*)wrow + hi;   // chunk k: bp[2*k] (b64 load)

    // prologue: fill both buffers
    async_stage(xbase, tileoff + 0 * (KB * 4), ldsb0, lane);
    async_stage(xbase, tileoff + 1 * (KB * 4), ldsb1, lane);

    v8f acc = {};

    for (int s = 0; s < NSTAGES - 2; ++s) {
        wait_async_le16();                       // stage s landed in LDS
        const int sel = s & 1;
        const v2f* ap = (const v2f*)(&xstage[wave][sel][m][0]) + hi;
        const v2f* bs = bp + (size_t)s * (2 * CPS);
        #pragma unroll
        for (int c = 0; c < CPS; ++c) {
            v2f a = ap[2 * c];                   // ds_load_b64, imm offset
            v2f b = bs[2 * c];                   // global_load_b64, imm offset
            acc = __builtin_amdgcn_wmma_f32_16x16x4_f32(
                false, a, false, b, (short)0, acc, false, false);
        }
        wait_ds_0();                             // buffer fully read
        async_stage(xbase, tileoff + (unsigned)(s + 2) * (KB * 4),
                    sel ? ldsb1 : ldsb0, lane);  // refill
    }

    // drain: stages NSTAGES-2 and NSTAGES-1
    #pragma unroll
    for (int t2 = 0; t2 < 2; ++t2) {
        const int s = NSTAGES - 2 + t2;
        if (t2 == 0) wait_async_le16(); else wait_async_0();
        const int sel = s & 1;
        const v2f* ap = (const v2f*)(&xstage[wave][sel][m][0]) + hi;
        const v2f* bs = bp + (size_t)s * (2 * CPS);
        #pragma unroll
        for (int c = 0; c < CPS; ++c) {
            v2f a = ap[2 * c];
            v2f b = bs[2 * c];
            acc = __builtin_amdgcn_wmma_f32_16x16x4_f32(
                false, a, false, b, (short)0, acc, false, false);
        }
    }

    // C layout: VGPR v, lanes 0-15 -> [M=v][N=lane]; lanes 16-31 -> [M=v+8][N-16]
    #pragma unroll
    for (int v = 0; v < 8; ++v)
        cscr[wave][v + hi * 8][m] = acc[v];

    __syncthreads();

    if (lane < 16) {
        const int t  = lane;             // token within tile
        const int gt = tile * 16 + t;    // global token

        const float4* srow = (const float4*)&cscr[wave][t][0];
        float4 g0 = srow[0], g1 = srow[1];   // gate logits e=0..7
        float4 n0 = srow[2], n1 = srow[3];   // noise pre-act e=0..7

        const float4* ep = (const float4*)(eps + (size_t)gt * RE);
        float4 e0 = ep[0], e1 = ep[1];

        float gate[8] = {g0.x, g0.y, g0.z, g0.w, g1.x, g1.y, g1.z, g1.w};
        float npre[8] = {n0.x, n0.y, n0.z, n0.w, n1.x, n1.y, n1.z, n1.w};
        float ee[8]   = {e0.x, e0.y, e0.z, e0.w, e1.x, e1.y, e1.z, e1.w};

        float v1 = -INFINITY, v2 = -INFINITY;
        int   i1 = 0, i2 = 0;
        #pragma unroll
        for (int e = 0; e < 8; ++e) {
            // numerically stable softplus: max(x,0) + log1p(exp(-|x|))
            float p  = npre[e];
            float sp = fmaxf(p, 0.0f) + log1pf(expf(-fabsf(p)));
            float lg = fmaf(sp, ee[e], gate[e]);
            if (lg > v1)      { v2 = v1; i2 = i1; v1 = lg; i1 = e; }
            else if (lg > v2) { v2 = lg; i2 = e; }
        }

        *(float2*)(out + (size_t)gt * 2) = make_float2(v1, v2);
        int* oidx = (int*)(out + (size_t)RTOK * 2);
        *(int2*)(oidx + (size_t)gt * 2) = make_int2(i1, i2);
    }
}

extern "C" void kernel_launch(void* const* d_in, const int* in_sizes, int n_in,
                              void* d_out, int out_size, void* d_ws, size_t ws_size,
                              hipStream_t stream) {
    const float* x       = (const float*)d_in[0];
    const float* w_g     = (const float*)d_in[1];
    const float* w_noise = (const float*)d_in[2];
    const float* eps     = (const float*)d_in[3];

    const int tiles  = RTOK / 16;                 // 1024
    const int blocks = tiles / WAVES_PER_BLOCK;   // 256
    noisy_topk_router_wmma<<<blocks, 128, 0, stream>>>(
        x, w_g, w_noise, eps, (float*)d_out);
}